// FullyConnectedCapsule_22857815949492
// MI455X (gfx1250) — compile-verified
//
#include <hip/hip_runtime.h>
#include <math.h>

// Problem constants (from reference): B=16384, N=32, J=16, I=8, D=16
constexpr int kN  = 32;   // output capsules (softmax axis)
constexpr int kJ  = 16;   // input capsules
constexpr int kI  = 8;    // input capsule dim
constexpr int kD  = 16;   // output capsule dim
constexpr int kTB = 16;   // batch rows per workgroup tile (WMMA M)
constexpr int kPitch = 130; // padded LDS row pitch (floats): 128 data + 2 pad -> bank-conflict-free

typedef float v2f __attribute__((ext_vector_type(2)));
typedef float v8f __attribute__((ext_vector_type(8)));
typedef unsigned int v4u __attribute__((ext_vector_type(4)));
typedef int v4i __attribute__((ext_vector_type(4)));
typedef int v8i __attribute__((ext_vector_type(8)));

// D = A(16x4 f32) * B(4x16 f32) + C(16x16 f32)
static __device__ __forceinline__ v8f wmma4(v2f a, v2f b, v8f c) {
  return __builtin_amdgcn_wmma_f32_16x16x4_f32(
      /*neg_a=*/false, a, /*neg_b=*/false, b,
      /*c_mod=*/(short)0, c, /*reuse_a=*/false, /*reuse_b=*/false);
}

// Horizontal add across the 16 lanes of each half-wave (z dimension of C/D layout).
// ds_swizzle group-of-32 mode: offset = xor_mask<<10 | and_mask(0x1F); xor<16 stays in-half.
#define HRED16(p)                                                                         \
  do {                                                                                    \
    p += __int_as_float(__builtin_amdgcn_ds_swizzle(__float_as_int(p), 0x041F));          \
    p += __int_as_float(__builtin_amdgcn_ds_swizzle(__float_as_int(p), 0x081F));          \
    p += __int_as_float(__builtin_amdgcn_ds_swizzle(__float_as_int(p), 0x101F));          \
    p += __int_as_float(__builtin_amdgcn_ds_swizzle(__float_as_int(p), 0x201F));          \
  } while (0)

__global__ __launch_bounds__(1024)
void caps_fc_kernel(const float* __restrict__ X,     // (B, J, I)
                    const float* __restrict__ W,     // (N, J, I, D)
                    const float* __restrict__ Bias,  // (N, J, 1)
                    float* __restrict__ Out) {       // (B, N, D)
  __shared__ __align__(16) float Xs[kTB * kPitch];   // staged X tile (padded pitch 130)
  __shared__ __align__(16) float Cs[kN * kJ * kTB];  // routing logits/weights, 32KB

  const int tid  = threadIdx.x;
  const int lane = tid & 31;
  const int n    = tid >> 5;        // wave id == output capsule index (32 waves)
  const int hh   = lane >> 4;       // half-wave
  const int lq   = lane & 15;       // lane within half
  const int bt   = blockIdx.x * kTB;

  // ---- stage X tile into LDS: 16 rows x 128 f32, hardware-padded to pitch 130 ----
#if __has_builtin(__builtin_amdgcn_tensor_load_to_lds)
  if (n == 0) {  // one wave issues the TDM descriptor; EXEC is ignored by tensor ops
    const unsigned long long gaddr =
        (unsigned long long)(const void*)(X + (size_t)bt * (kJ * kI));
    const unsigned int ldsa = (unsigned int)(size_t)&Xs[0];

    // D# group 0: count=1 | lds_addr | global_addr[56:0] | type=2
    v4u g0;
    g0.x = 1u;                                  // count=1, is_restore=0, gather off
    g0.y = ldsa;                                // lds_addr (bytes)
    g0.z = (unsigned int)gaddr;                 // global_addr[31:0]
    g0.w = (unsigned int)((gaddr >> 32) & 0x01FFFFFFu) | (2u << 30);  // [56:32] | type=2

    // D# group 1: data_size=4B, pad after every 128 DWORDs by 2 DWORDs (pitch 130),
    // tensor_dim0=128, tensor_dim1=16, tile_dim0=128, tile_dim1=16, dim0_stride=128
    unsigned int w0 = (2u << 16)      // data_size = 4 bytes
                    | (1u << 20)      // pad_enable
                    | (6u << 22)      // pad_interval: 2^(6+1)=128 DWORDs
                    | (1u << 25);     // pad_amount: 2 DWORDs
    unsigned int w1 = (128u << 16);   // tensor_dim0[15:0] -> bits[63:48]
    unsigned int w2 = (16u << 16);    // tensor_dim0[31:16]=0 | tensor_dim1[15:0]=16
    unsigned int w3 = (128u << 16);   // tensor_dim1[31:16]=0 | tile_dim0=128
    unsigned int w4 = 16u;            // tile_dim1=16 | tile_dim2=0
    unsigned int w5 = 128u;           // tensor_dim0_stride[31:0]
    unsigned int w6 = 0u;             // stride[47:32]=0 | tensor_dim1_stride[15:0]=0
    unsigned int w7 = 0u;
    v8i g1 = {(int)w0, (int)w1, (int)w2, (int)w3, (int)w4, (int)w5, (int)w6, (int)w7};

    v4i gz = {0, 0, 0, 0};            // groups 2/3: unused dims for a 2D tile
#if defined(__clang_major__) && (__clang_major__ >= 23)
    v8i gz8 = {0, 0, 0, 0, 0, 0, 0, 0};
    __builtin_amdgcn_tensor_load_to_lds(g0, g1, gz, gz, gz8, 0);
#else
    __builtin_amdgcn_tensor_load_to_lds(g0, g1, gz, gz, 0);
#endif
    __builtin_amdgcn_s_wait_tensorcnt(0);
  }
#else
  {
    const float2* src = (const float2*)(X + (size_t)bt * (kJ * kI));
    float2 v = src[tid];
    const int e = tid * 2;
    const int b = e >> 7;
    const int f = e & 127;
    *(float2*)&Xs[b * kPitch + f] = v;
  }
#endif

  // ---- preload W[n] as WMMA B-tiles: Wb[j][t], K-chunk t covers i = 4t..4t+3 ----
  // B layout: VGPR r, lane l holds W[K = r + 2*(l>=16), N = l%16]
  v2f Wb[kJ][2];
  const float* Wn = W + (size_t)n * (kJ * kI * kD);
  __builtin_prefetch(Wn, 0, 0);
#pragma unroll
  for (int j = 0; j < kJ; ++j) {
#pragma unroll
    for (int t = 0; t < 2; ++t) {
      const int i0 = t * 4 + 2 * hh;
      Wb[j][t].x = Wn[j * (kI * kD) + (i0    ) * kD + lq];
      Wb[j][t].y = Wn[j * (kI * kD) + (i0 + 1) * kD + lq];
    }
  }

  __syncthreads();

  // ---- pass 1a: Ssum[b,z] = sum_{j,i} X[b,j,i] * W[n,j,i,z]  (one chained K=128 GEMM) ----
  v8f Ssum = {0.f, 0.f, 0.f, 0.f, 0.f, 0.f, 0.f, 0.f};
#pragma unroll
  for (int j = 0; j < kJ; ++j) {
    const float* xr = &Xs[lq * kPitch + j * kI + 2 * hh];  // A tile: M=lq, K pair = consecutive i
    const v2f A0 = *(const v2f*)(xr);       // i = 2*hh, 2*hh+1
    const v2f A1 = *(const v2f*)(xr + 4);   // i = 4+2*hh, 5+2*hh
    Ssum = wmma4(A0, Wb[j][0], Ssum);
    Ssum = wmma4(A1, Wb[j][1], Ssum);
  }

  // ---- pass 1b: per j recompute u_j, logits c[b,j] = (1/sqrt(D)) * sum_z u_j[b,z]*Ssum[b,z] ----
#pragma unroll
  for (int j = 0; j < kJ; ++j) {
    const float* xr = &Xs[lq * kPitch + j * kI + 2 * hh];
    const v2f A0 = *(const v2f*)(xr);
    const v2f A1 = *(const v2f*)(xr + 4);
    v8f acc = {0.f, 0.f, 0.f, 0.f, 0.f, 0.f, 0.f, 0.f};
    acc = wmma4(A0, Wb[j][0], acc);
    acc = wmma4(A1, Wb[j][1], acc);
    // C/D layout: VGPR r, lane l -> u[b = r + 8*(l>=16), z = l%16]
#pragma unroll
    for (int r = 0; r < 8; ++r) {
      float p = acc[r] * Ssum[r];
      HRED16(p);                                   // replicated across each half
      if (lq == 0)
        Cs[n * (kJ * kTB) + j * kTB + (r + 8 * hh)] = p * 0.25f;  // /sqrt(16)
    }
  }

  __syncthreads();

  // ---- softmax over n (axis=1) for each (j, b), then add bias[n,j] ----
  if (tid < kJ * kTB) {
    const int jj = tid >> 4;
    const int bb = tid & 15;
    const int base = jj * kTB + bb;
    float e[kN];
    float m = -3.402823466e38f;
#pragma unroll
    for (int k = 0; k < kN; ++k) {
      const float v = Cs[k * (kJ * kTB) + base];
      e[k] = v;
      m = fmaxf(m, v);
    }
    float s = 0.f;
#pragma unroll
    for (int k = 0; k < kN; ++k) { e[k] = __expf(e[k] - m); s += e[k]; }
    const float inv = 1.0f / s;
#pragma unroll
    for (int k = 0; k < kN; ++k)
      Cs[k * (kJ * kTB) + base] = e[k] * inv + Bias[k * kJ + jj];
  }

  __syncthreads();

  // ---- pass 2: s[b,z] = sum_{j,i} (c[b,j]*X[b,j,i]) * W[n,j,i,z]  (second K=128 GEMM) ----
  v8f S = {0.f, 0.f, 0.f, 0.f, 0.f, 0.f, 0.f, 0.f};
#pragma unroll
  for (int j = 0; j < kJ; ++j) {
    const float cj = Cs[n * (kJ * kTB) + j * kTB + lq];    // A tile M = lq = b
    const float* xr = &Xs[lq * kPitch + j * kI + 2 * hh];
    v2f A0, A1;
    A0.x = xr[0] * cj; A0.y = xr[1] * cj;
    A1.x = xr[4] * cj; A1.y = xr[5] * cj;
    S = wmma4(A0, Wb[j][0], S);
    S = wmma4(A1, Wb[j][1], S);
  }

  // ---- store s tile: Out[b, n, z], C/D layout row b = r + 8*hh, col z = lq ----
#pragma unroll
  for (int r = 0; r < 8; ++r) {
    const int b = bt + r + 8 * hh;
    Out[(size_t)b * (kN * kD) + n * kD + lq] = S[r];
  }
}

extern "C" void kernel_launch(void* const* d_in, const int* in_sizes, int n_in,
                              void* d_out, int out_size, void* d_ws, size_t ws_size,
                              hipStream_t stream) {
  const float* X    = (const float*)d_in[0];  // (B, 16, 8)
  const float* W    = (const float*)d_in[1];  // (32, 16, 8, 16)
  const float* Bias = (const float*)d_in[2];  // (32, 16, 1)
  float* Out        = (float*)d_out;          // (B, 32, 16)

  const int B = in_sizes[0] / (kJ * kI);      // 16384
  const int grid = B / kTB;                   // 1024 workgroups, 16 batch rows each

  caps_fc_kernel<<<grid, kN * 32, 0, stream>>>(X, W, Bias, Out);
}